// DirectInterpGNN_91096256348925
// MI455X (gfx1250) — compile-verified
//
#include <hip/hip_runtime.h>
#include <hip/hip_bf16.h>
#include <stdint.h>

// ---------------- configuration ----------------
#define TILE        1024                 // edges per LDS tile
#define TB          256                  // threads per block (8 wave32)
#define ATTR_BYTES  (TILE * 12)          // 12 KB of edge_attr per tile
#define IDX_BYTES   (TILE * 4)           // 4 KB of src indices per tile
#define BUF_BYTES   (ATTR_BYTES + IDX_BYTES)
#define BLOCKS_ACC  2048

typedef float v4f __attribute__((ext_vector_type(4)));
typedef int   v4i __attribute__((ext_vector_type(4)));

// Generic shared pointer: low 32 bits are the LDS byte offset.
__device__ __forceinline__ unsigned lds_off(const void* p) {
  return (unsigned)(unsigned long long)(uintptr_t)p;
}

// gfx1250 async DMA: global -> LDS, 16B per lane, tracked by ASYNCcnt.
__device__ __forceinline__ void async_copy_b128(unsigned lds_byte, const void* gsrc) {
  asm volatile("global_load_async_to_lds_b128 %0, %1, off"
               :: "v"(lds_byte), "v"(gsrc) : "memory");
}

__device__ __forceinline__ void atomic_add_f32(float* p, float v) {
  // relaxed, agent scope -> should lower to global_atomic_add_f32 (no return)
  __hip_atomic_fetch_add(p, v, __ATOMIC_RELAXED, __HIP_MEMORY_SCOPE_AGENT);
}

// ---------------- kernel: scatter-accumulate (async LDS pipeline) --------
// First in the file so the disasm snippet shows the async-copy + atomic
// lowering for verification.
__global__ void gnn_accum_kernel(const float* __restrict__ edge_attr,
                                 const int* __restrict__ src,
                                 float* __restrict__ nd,   // [2*V] interleaved num,den
                                 long long E) {
  __shared__ __align__(16) unsigned char smem[2][BUF_BYTES];
  const int tid = threadIdx.x;
  const long long fullTiles = E / TILE;
  const long long stride = gridDim.x;
  const char* attr_base = (const char*)edge_attr;
  const char* idx_base  = (const char*)src;
  const unsigned lbuf0 = lds_off(&smem[0][0]);
  const unsigned lbuf1 = lds_off(&smem[1][0]);

  // Issue one tile's copies: 3 attr chunks + 1 idx chunk of 16B per thread
  // => exactly 4 async instructions per wave per tile.
  auto issue = [&](int b, long long t) {
    const char* ag = attr_base + t * (long long)ATTR_BYTES;
    const char* ig = idx_base  + t * (long long)IDX_BYTES;
    unsigned lb = b ? lbuf1 : lbuf0;
    async_copy_b128(lb + 16u * (unsigned)tid,            ag + 16 * (tid));
    async_copy_b128(lb + 16u * (unsigned)(tid + TB),     ag + 16 * (tid + TB));
    async_copy_b128(lb + 16u * (unsigned)(tid + 2 * TB), ag + 16 * (tid + 2 * TB));
    async_copy_b128(lb + (unsigned)ATTR_BYTES + 16u * (unsigned)tid, ig + 16 * (tid));
  };

  long long tile = blockIdx.x;
  int cur = 0;
  if (tile < fullTiles) issue(cur, tile);

  for (; tile < fullTiles; tile += stride) {
    const long long nxt = tile + stride;
    const bool has_next = nxt < fullTiles;
    if (has_next) {
      issue(cur ^ 1, nxt);
      // 4 ops outstanding for the next tile; async loads complete in order,
      // so waiting down to 4 means the current tile is fully in LDS.
      asm volatile("s_wait_asynccnt 0x4" ::: "memory");
    } else {
      asm volatile("s_wait_asynccnt 0x0" ::: "memory");
    }
    __syncthreads();  // publish all waves' LDS tiles

    const float* attr = (const float*)(&smem[cur][0]);
    const int*   sidx = (const int*)(&smem[cur][ATTR_BYTES]);
#pragma unroll
    for (int k = 0; k < TILE / TB; ++k) {
      const int e = tid + k * TB;
      const float A  = attr[3 * e + 0];
      const float S  = attr[3 * e + 1];
      const float vv = attr[3 * e + 2];
      const int s = sidx[e];
      atomic_add_f32(&nd[2 * s + 0], A);
      atomic_add_f32(&nd[2 * s + 1], A * S * vv);
    }
    __syncthreads();  // everyone done reading before buffer is overwritten
    cur ^= 1;
  }

  // Tail edges (E % TILE), direct path.
  const long long e = fullTiles * TILE + (long long)blockIdx.x * TB + tid;
  if (e < E) {
    const float A  = edge_attr[3 * e + 0];
    const float S  = edge_attr[3 * e + 1];
    const float vv = edge_attr[3 * e + 2];
    const int s = src[e];
    atomic_add_f32(&nd[2 * s + 0], A);
    atomic_add_f32(&nd[2 * s + 1], A * S * vv);
  }
}

// ---------------- kernel: zero the num/den accumulators ------------------
__global__ void gnn_zero_kernel(float* __restrict__ p, int n) {
  int i = blockIdx.x * blockDim.x + threadIdx.x;
  if (i < n) p[i] = 0.0f;
}

// ---------------- kernel: per-vertex t = (1-C) * num / (den * A_ii) ------
__global__ void gnn_vertex_kernel(const float2* __restrict__ vattr,
                                  float* __restrict__ nd, int V) {
  int v = blockIdx.x * blockDim.x + threadIdx.x;
  if (v < V) {
    const float num = nd[2 * v + 0];
    const float den = nd[2 * v + 1];
    const float2 a = vattr[v];               // a.x = A_ii, a.y = C_i
    const float t = (1.0f - a.y) * (num / (den * a.x));
    nd[2 * v + 0] = t;                        // in-place: own slot only
  }
}

// ---------------- kernel: w_ij = -A_ij * t[src] (vectorized, NT) ---------
__global__ void gnn_finalize_kernel(const float* __restrict__ edge_attr,
                                    const int* __restrict__ src,
                                    const float* __restrict__ nd,
                                    float* __restrict__ w,
                                    long long E) {
  const long long g = (long long)blockIdx.x * blockDim.x + threadIdx.x;
  const long long G = E / 4;  // groups of 4 edges -> pure b128 traffic
  if (g < G) {
    const v4f* attr4 = (const v4f*)edge_attr;
    const v4i* src4  = (const v4i*)src;
    v4f a0 = __builtin_nontemporal_load(&attr4[3 * g + 0]);
    v4f a1 = __builtin_nontemporal_load(&attr4[3 * g + 1]);
    v4f a2 = __builtin_nontemporal_load(&attr4[3 * g + 2]);
    v4i s  = __builtin_nontemporal_load(&src4[g]);
    // A_ij sits at flat lanes 0,3,6,9 of the 12 packed floats
    v4f r;
    r.x = -a0.x * nd[2 * s.x];
    r.y = -a0.w * nd[2 * s.y];
    r.z = -a1.z * nd[2 * s.z];
    r.w = -a2.y * nd[2 * s.w];
    __builtin_nontemporal_store(r, &((v4f*)w)[g]);
  }
  // Tail (E % 4 edges) handled by the first few threads of the grid.
  const long long rem = E - G * 4;
  if (g < rem) {
    const long long e = G * 4 + g;
    const float A = edge_attr[3 * e];
    const int s = src[e];
    w[e] = -A * nd[2 * s];
  }
}

// ---------------- launcher ----------------
extern "C" void kernel_launch(void* const* d_in, const int* in_sizes, int n_in,
                              void* d_out, int out_size, void* d_ws, size_t ws_size,
                              hipStream_t stream) {
  const float* vattr = (const float*)d_in[0];   // (V,2)
  const float* eattr = (const float*)d_in[1];   // (E,3)
  const int*   epair = (const int*)d_in[2];     // (2,E); row 0 = src
  const int V = in_sizes[0] / 2;
  const long long E = (long long)in_sizes[1] / 3;
  const int* src = epair;

  float* nd = (float*)d_ws;      // 2*V floats = 4 MB scratch
  float* w  = (float*)d_out;     // E floats

  const int nzero = 2 * V;
  gnn_zero_kernel<<<(nzero + TB - 1) / TB, TB, 0, stream>>>(nd, nzero);

  gnn_accum_kernel<<<BLOCKS_ACC, TB, 0, stream>>>(eattr, src, nd, E);

  gnn_vertex_kernel<<<(V + TB - 1) / TB, TB, 0, stream>>>((const float2*)vattr, nd, V);

  long long G = E / 4;
  int fblocks = (int)((G + TB - 1) / TB);
  if (fblocks < 1) fblocks = 1;
  gnn_finalize_kernel<<<fblocks, TB, 0, stream>>>(eattr, src, nd, w, E);
}